// PatchMatching_19619410608194
// MI455X (gfx1250) — compile-verified
//
#include <hip/hip_runtime.h>
#include <hip/hip_bf16.h>
#include <math.h>

typedef __attribute__((ext_vector_type(2))) float v2f;
typedef __attribute__((ext_vector_type(8))) float v8f;

#define IN_W   384
#define IN_HW  (384*384)      // 147456
#define W      96
#define HW     (96*96)        // 9216
#define NPLANE 72             // 54 census planes + 18 raw planes
#define NBLK   (2*96*6)       // 1152 pixel-tile blocks

// ---------------- helpers ----------------

// reflect (no edge repeat) on [0,95]; valid for overshoot <= 95
__device__ __forceinline__ int refl96(int i) {
    return i < 0 ? -i : (i > 95 ? 190 - i : i);
}

// Keys cubic kernel, a = -0.5 (jax.image 'cubic')
__device__ __forceinline__ float keys(float x) {
    x = fabsf(x);
    if (x <= 1.f) return ((1.5f * x - 2.5f) * x) * x + 1.f;
    if (x <  2.f) return ((-0.5f * x + 2.5f) * x - 4.f) * x + 2.f;
    return 0.f;
}
// antialiased weight for tap d (0..15), kernel stretched by scale 4,
// output sample sits at fractional offset +7.5 from tap base (4*o - 6)
__device__ __forceinline__ float dsw(int d) {
    return keys(((float)d - 7.5f) * 0.25f);
}

// ---------------- kernel 1: grayscale*255 for 6 images ----------------
__global__ void gray_kernel(const float* __restrict__ pred,
                            const float* __restrict__ I,
                            float* __restrict__ gray) {
    int idx = blockIdx.x * blockDim.x + threadIdx.x;
    if (idx >= 6 * IN_HW) return;
    int pix = idx % IN_HW;
    int gi  = idx / IN_HW;             // 0,1: pred b ; 2,3: I f=0 b ; 4,5: I f=1 b
    const float* base;
    if (gi < 2) {
        base = pred + (size_t)gi * 3 * IN_HW;
    } else {
        int f = (gi - 2) / 2;
        int b = (gi - 2) % 2;
        base = I + ((size_t)(b * 2 + f) * 3) * IN_HW;
    }
    float r  = base[pix];
    float g  = base[IN_HW + pix];
    float bl = base[2 * IN_HW + pix];
    gray[idx] = (0.299f * r + 0.587f * g + 0.114f * bl) * 255.0f;
}

// ---------------- kernel 2: build 72 source planes (census + raw copy) ----------
__global__ void prep_kernel(const float* __restrict__ pred,
                            const float* __restrict__ I,
                            const float* __restrict__ gray,
                            float* __restrict__ S) {
    int idx = blockIdx.x * blockDim.x + threadIdx.x;
    if (idx >= NPLANE * IN_HW) return;
    int pix = idx % IN_HW;
    int p   = idx / IN_HW;
    float out;
    if (p < 54) {
        // census planes: 0..17 pred(b,c), 18..35 f=0(b,c), 36..53 f=1(b,c)
        int gi, ch = p % 9;
        if (p < 18)      gi = p / 9;
        else if (p < 36) gi = 2 + (p - 18) / 9;
        else             gi = 4 + (p - 36) / 9;
        const float* g = gray + (size_t)gi * IN_HW;
        int y = pix / IN_W, x = pix % IN_W;
        int oy = ch / 3 - 1, ox = ch % 3 - 1;
        int yy = y + oy, xx = x + ox;
        float nb = (yy >= 0 && yy < IN_W && xx >= 0 && xx < IN_W) ? g[yy * IN_W + xx] : 0.f;
        float t = nb - g[pix];
        out = t / sqrtf(0.81f + t * t);
    } else {
        // raw planes: 54..59 pred(b,c), 60..65 I f=0, 66..71 I f=1
        int q = p - 54;
        const float* src;
        if (q < 6) {
            src = pred + (size_t)q * IN_HW;            // q == b*3+c
        } else if (q < 12) {
            int b = (q - 6) / 3, c = (q - 6) % 3;
            src = I + ((size_t)(b * 2 + 0) * 3 + c) * IN_HW;
        } else {
            int b = (q - 12) / 3, c = (q - 12) % 3;
            src = I + ((size_t)(b * 2 + 1) * 3 + c) * IN_HW;
        }
        out = src[pix];
    }
    S[idx] = out;
}

// ---------------- kernel 3: vertical bicubic downsample 384 -> 96 --------------
__global__ void dsy_kernel(const float* __restrict__ S, float* __restrict__ M) {
    int idx = blockIdx.x * blockDim.x + threadIdx.x;
    if (idx >= NPLANE * 96 * IN_W) return;
    int x  = idx % IN_W;
    int oy = (idx / IN_W) % 96;
    int p  = idx / (IN_W * 96);
    const float* src = S + (size_t)p * IN_HW;
    int n0 = 4 * oy - 6;
    float acc = 0.f, wsum = 0.f;
#pragma unroll
    for (int d = 0; d < 16; ++d) {
        int n = n0 + d;
        if (n >= 0 && n < IN_W) {
            float w = dsw(d);
            acc  += w * src[(size_t)n * IN_W + x];
            wsum += w;
        }
    }
    M[idx] = acc / wsum;      // per-row renormalization == jax boundary behavior
}

// ---------------- kernel 4: horizontal bicubic downsample 384 -> 96 ------------
__global__ void dsx_kernel(const float* __restrict__ M, float* __restrict__ D) {
    int idx = blockIdx.x * blockDim.x + threadIdx.x;
    if (idx >= NPLANE * HW) return;
    int ox = idx % W;
    int y  = (idx / W) % 96;
    int p  = idx / HW;
    const float* row = M + ((size_t)p * 96 + y) * IN_W;
    int n0 = 4 * ox - 6;
    float acc = 0.f, wsum = 0.f;
#pragma unroll
    for (int d = 0; d < 16; ++d) {
        int n = n0 + d;
        if (n >= 0 && n < IN_W) {
            float w = dsw(d);
            acc  += w * row[n];
            wsum += w;
        }
    }
    D[idx] = acc / wsum;
}

// ---------------- kernel 5: per-pixel squared 81-dim patch norms ---------------
__global__ void sq_kernel(const float* __restrict__ D, float* __restrict__ SQ) {
    int idx = blockIdx.x * blockDim.x + threadIdx.x;
    if (idx >= 6 * HW) return;
    int pix = idx % HW;
    int mi  = idx / HW;                // 0,1 pred(b) ; 2,3 f0(b) ; 4,5 f1(b)
    int base;
    if (mi < 2)      base = mi * 9;
    else if (mi < 4) base = 18 + (mi - 2) * 9;
    else             base = 36 + (mi - 4) * 9;
    int y = pix / W, x = pix % W;
    float acc = 0.f;
#pragma unroll
    for (int c = 0; c < 9; ++c)
#pragma unroll
        for (int j = 0; j < 9; ++j) {
            int oy = j / 3 - 1, ox = j % 3 - 1;
            float v = D[(size_t)(base + c) * HW + refl96(y + oy) * W + refl96(x + ox)];
            acc += v * v;
        }
    SQ[idx] = acc;
}

// ---------------- kernel 6: WMMA cost volume + argmin + gathered MSE -----------
__global__ void __launch_bounds__(32)
cost_argmin_loss_kernel(const float* __restrict__ D,
                        const float* __restrict__ SQ,
                        float* __restrict__ partials) {
    __shared__ float Alds[16 * 84];    // 16 pixels x 84 (81 census feats, pad 0)
    __shared__ float Blds[22 * 84];    // 22 shifted columns x 84
    __shared__ float Clds[16 * 22];    // dot products

    const int lane = threadIdx.x;
    int blk = blockIdx.x;              // b*576 + y*6 + xt
    const int xt = blk % 6;  blk /= 6;
    const int y  = blk % 96; blk /= 96;
    const int b  = blk;
    const int x0 = xt * 16;

    const float* censP = D + (size_t)(b * 9) * HW;

    // stage A tile (pred census patch vectors for 16 pixels in row y)
    for (int i = lane; i < 16 * 84; i += 32) {
        const int m = i / 84, k = i % 84;
        float v = 0.f;
        if (k < 81) {
            const int c = k / 9, j = k % 9;
            const int oy = j / 3 - 1, ox = j % 3 - 1;
            v = censP[(size_t)c * HW + refl96(y + oy) * W + refl96(x0 + m + ox)];
        }
        Alds[i] = v;
    }
    __syncthreads();

    const int m16  = lane & 15;
    const int koff = (lane < 16) ? 0 : 2;   // K-half per WMMA f32 16x16x4 layout

    // preload the 21 A fragments (K = 84 / 4) once, reused for all 14 (f,dy)
    v2f afrag[21];
#pragma unroll
    for (int s = 0; s < 21; ++s) {
        afrag[s].x = Alds[m16 * 84 + 4 * s + koff];
        afrag[s].y = Alds[m16 * 84 + 4 * s + koff + 1];
    }

    const float p2 = SQ[(size_t)b * HW + y * W + x0 + m16];

    float best   = 3.4e38f;
    int  bestIdx = 0;

    for (int f = 0; f < 2; ++f) {
        const float* censF = D  + (size_t)(18 + f * 18 + b * 9) * HW;
        const float* sqF   = SQ + (size_t)(2 + f * 2 + b) * HW;
        for (int dy = -3; dy <= 3; ++dy) {
            const int yy = refl96(y + dy);
            __syncthreads();   // previous Clds reads / Blds reads done
            // stage B tile: census patch vectors at row yy, cols x0-3 .. x0+18
            for (int i = lane; i < 22 * 84; i += 32) {
                const int col = i / 84, k = i % 84;
                float v = 0.f;
                if (k < 81) {
                    const int c = k / 9, j = k % 9;
                    const int oy = j / 3 - 1, ox = j % 3 - 1;
                    const int xc = refl96(x0 - 3 + col);
                    v = censF[(size_t)c * HW + refl96(yy + oy) * W + refl96(xc + ox)];
                }
                Blds[i] = v;
            }
            __syncthreads();

            v8f acc1 = {0.f, 0.f, 0.f, 0.f, 0.f, 0.f, 0.f, 0.f};
            v8f acc2 = acc1;
            const int col2 = (16 + m16 < 22) ? (16 + m16) : 21;  // clamp padding cols
#pragma unroll
            for (int s = 0; s < 21; ++s) {
                v2f bf1, bf2;
                bf1.x = Blds[m16 * 84 + 4 * s + koff];
                bf1.y = Blds[m16 * 84 + 4 * s + koff + 1];
                bf2.x = Blds[col2 * 84 + 4 * s + koff];
                bf2.y = Blds[col2 * 84 + 4 * s + koff + 1];
                acc1 = __builtin_amdgcn_wmma_f32_16x16x4_f32(
                    false, afrag[s], false, bf1, (short)0, acc1, false, false);
                acc2 = __builtin_amdgcn_wmma_f32_16x16x4_f32(
                    false, afrag[s], false, bf2, (short)0, acc2, false, false);
            }

            // scatter C tiles: lane n holds column n, rows v (+8 for upper lanes)
            const int rowOff = (lane < 16) ? 0 : 8;
#pragma unroll
            for (int v = 0; v < 8; ++v) {
                Clds[(v + rowOff) * 22 + m16] = acc1[v];
                if (m16 < 6) Clds[(v + rowOff) * 22 + 16 + m16] = acc2[v];
            }
            __syncthreads();

            if (lane < 16) {
                const int m = lane;
#pragma unroll
                for (int dx = -3; dx <= 3; ++dx) {
                    const int n   = m + dx + 3;           // column holding (m,dx)
                    const float dot = Clds[m * 22 + n];
                    const int xx  = refl96(x0 + m + dx);
                    const float dis = p2 + sqF[yy * W + xx] - 2.0f * dot;
                    const int idx = f * 49 + (dy + 3) * 7 + (dx + 3);
                    if (dis < best) { best = dis; bestIdx = idx; } // first-wins ties
                }
            }
        }
    }

    // gathered raw-patch MSE for this pixel (lanes 0..15)
    float part = 0.f;
    if (lane < 16) {
        const int m  = lane;
        const int f  = bestIdx / 49;
        const int r  = bestIdx % 49;
        const int dy = r / 7 - 3, dx = r % 7 - 3;
        const int qy = refl96(y + dy), qx = refl96(x0 + m + dx);
        const float* rawP = D + (size_t)(54 + b * 3) * HW;
        const float* rawF = D + (size_t)(60 + f * 6 + b * 3) * HW;
        float acc = 0.f;
#pragma unroll
        for (int c = 0; c < 3; ++c)
#pragma unroll
            for (int j = 0; j < 9; ++j) {
                const int oy = j / 3 - 1, ox = j % 3 - 1;
                float pv = rawP[(size_t)c * HW + refl96(y + oy)  * W + refl96(x0 + m + ox)];
                float mv = rawF[(size_t)c * HW + refl96(qy + oy) * W + refl96(qx + ox)];
                float d  = pv - mv;
                acc += d * d;
            }
        part = 0.5f * acc;
    }
    // wave32 reduction
#pragma unroll
    for (int off = 16; off >= 1; off >>= 1)
        part += __shfl_xor(part, off, 32);
    if (lane == 0) partials[blockIdx.x] = part;
}

// ---------------- kernel 7: deterministic final reduction ----------------------
__global__ void reduce_kernel(const float* __restrict__ partials, float* __restrict__ out) {
    __shared__ float sh[256];
    float s = 0.f;
    for (int i = threadIdx.x; i < NBLK; i += 256) s += partials[i];
    sh[threadIdx.x] = s;
    __syncthreads();
    for (int o = 128; o >= 1; o >>= 1) {
        if (threadIdx.x < (unsigned)o) sh[threadIdx.x] += sh[threadIdx.x + o];
        __syncthreads();
    }
    if (threadIdx.x == 0) out[0] = sh[0] / 497664.0f;  // 2 * 9216 * 27 elements
}

// ---------------- host launcher ------------------------------------------------
extern "C" void kernel_launch(void* const* d_in, const int* in_sizes, int n_in,
                              void* d_out, int out_size, void* d_ws, size_t ws_size,
                              hipStream_t stream) {
    (void)in_sizes; (void)n_in; (void)out_size; (void)ws_size;
    const float* pred = (const float*)d_in[0];     // [2,3,384,384]
    const float* I    = (const float*)d_in[1];     // [2,2,3,384,384]
    float* out = (float*)d_out;
    float* ws  = (float*)d_ws;

    // workspace layout (floats)
    float* gray = ws;                               //  6 * 147456
    float* S    = gray + (size_t)6 * IN_HW;         // 72 * 147456
    float* MID  = S    + (size_t)NPLANE * IN_HW;    // 72 * 96 * 384
    float* D    = MID  + (size_t)NPLANE * 96 * IN_W;// 72 * 9216
    float* SQ   = D    + (size_t)NPLANE * HW;       //  6 * 9216
    float* PART = SQ   + (size_t)6 * HW;            // 1152

    const int T = 256;
    gray_kernel<<<(6 * IN_HW + T - 1) / T, T, 0, stream>>>(pred, I, gray);
    prep_kernel<<<(NPLANE * IN_HW + T - 1) / T, T, 0, stream>>>(pred, I, gray, S);
    dsy_kernel <<<(NPLANE * 96 * IN_W + T - 1) / T, T, 0, stream>>>(S, MID);
    dsx_kernel <<<(NPLANE * HW + T - 1) / T, T, 0, stream>>>(MID, D);
    sq_kernel  <<<(6 * HW + T - 1) / T, T, 0, stream>>>(D, SQ);
    cost_argmin_loss_kernel<<<NBLK, 32, 0, stream>>>(D, SQ, PART);
    reduce_kernel<<<1, 256, 0, stream>>>(PART, out);
}